// DGC_32847909880137
// MI455X (gfx1250) — compile-verified
//
#include <hip/hip_runtime.h>
#include <hip/hip_bf16.h>
#include <float.h>

#define BATCH   8
#define NPTS    2048
#define CH      128     // feature channels C
#define HALF_CH 64      // conv output channels per branch
#define KNN_K   20
#define EPSB    1e-5f

typedef __attribute__((ext_vector_type(2))) float v2f;
typedef __attribute__((ext_vector_type(8))) float v8f;

// ---------------------------------------------------------------------------
// Kernel 1: fold BN into weights/biases and build the combined 128x128 weight
//   Wc rows 0..63   = s2 * W2a          (neighbor term)
//   Wc rows 64..127 = s2 * (W2b - W2a)  (center term)
// plus the tiny geometry weights (64x3 each) and folded biases.
// ---------------------------------------------------------------------------
__global__ void prep_params(const float* __restrict__ w1, const float* __restrict__ b1,
                            const float* __restrict__ g1, const float* __restrict__ be1,
                            const float* __restrict__ m1, const float* __restrict__ v1,
                            const float* __restrict__ w2, const float* __restrict__ b2,
                            const float* __restrict__ g2, const float* __restrict__ be2,
                            const float* __restrict__ m2, const float* __restrict__ v2,
                            float* __restrict__ Wc,   float* __restrict__ sw1a,
                            float* __restrict__ sw1d, float* __restrict__ bias1p,
                            float* __restrict__ bias2p) {
  int o = threadIdx.x;
  if (o >= HALF_CH) return;
  float s1 = g1[o] / sqrtf(v1[o] + EPSB);
  bias1p[o] = b1[o] * s1 + (be1[o] - m1[o] * s1);
  #pragma unroll
  for (int c = 0; c < 3; ++c) {
    float wa = w1[o * 6 + c];
    float wb = w1[o * 6 + 3 + c];
    sw1a[o * 3 + c] = s1 * wa;
    sw1d[o * 3 + c] = s1 * (wb - wa);
  }
  float s2 = g2[o] / sqrtf(v2[o] + EPSB);
  bias2p[o] = b2[o] * s2 + (be2[o] - m2[o] * s2);
  for (int c = 0; c < CH; ++c) {
    float wa = w2[o * (2 * CH) + c];
    float wb = w2[o * (2 * CH) + CH + c];
    Wc[(size_t)o * CH + c]             = s2 * wa;
    Wc[(size_t)(HALF_CH + o) * CH + c] = s2 * (wb - wa);
  }
}

// ---------------------------------------------------------------------------
// Kernel 2: kNN. Whole batch's points (+squared norms) staged in LDS (32 KB of
// 320 KB). One thread per query row, broadcast LDS scan, register top-20.
// d = |xn|^2 + |xm|^2 - 2 xn.xm  (matches reference formula); strict '<'
// insertion keeps the earliest index on ties, matching jax.lax.top_k.
// ---------------------------------------------------------------------------
__global__ void knn_kernel(const float* __restrict__ xyz, int* __restrict__ idx_out) {
  __shared__ float4 pts[NPTS];
  const int b = blockIdx.y;
  const float* xb = xyz + (size_t)b * 3 * NPTS;
  for (int m = threadIdx.x; m < NPTS; m += blockDim.x) {
    float x = xb[m], y = xb[NPTS + m], z = xb[2 * NPTS + m];
    pts[m] = make_float4(x, y, z, x * x + y * y + z * z);
  }
  __syncthreads();

  const int n = blockIdx.x * blockDim.x + threadIdx.x;
  const float4 me = pts[n];

  float td[KNN_K];
  int   ti[KNN_K];
  #pragma unroll
  for (int i = 0; i < KNN_K; ++i) { td[i] = FLT_MAX; ti[i] = 0; }

  for (int m = 0; m < NPTS; ++m) {
    float4 p = pts[m];
    float dot = me.x * p.x + me.y * p.y + me.z * p.z;
    float d = me.w + p.w - 2.0f * dot;
    if (d < td[KNN_K - 1]) {
      td[KNN_K - 1] = d; ti[KNN_K - 1] = m;
      #pragma unroll
      for (int i = KNN_K - 1; i > 0; --i) {
        if (td[i] < td[i - 1]) {
          float tf = td[i]; td[i] = td[i - 1]; td[i - 1] = tf;
          int   tt = ti[i]; ti[i] = ti[i - 1]; ti[i - 1] = tt;
        }
      }
    }
  }
  int* op = idx_out + ((size_t)b * NPTS + n) * KNN_K;
  #pragma unroll
  for (int i = 0; i < KNN_K; ++i) op[i] = ti[i];
}

// ---------------------------------------------------------------------------
// Kernel 3: geometry transform. PQ[pt][0..63] = sW1a*xyz (neighbor term),
// PQ[pt][64..127] = sW1d*xyz + bias1' (center term). 64 channel-threads per
// point -> coalesced point-major stores.
// ---------------------------------------------------------------------------
__global__ void geo_kernel(const float* __restrict__ xyz, const float* __restrict__ sw1a,
                           const float* __restrict__ sw1d, const float* __restrict__ bias1p,
                           float* __restrict__ PQ) {
  int t  = blockIdx.x * blockDim.x + threadIdx.x;
  int ch = t & (HALF_CH - 1);
  int pt = t >> 6;
  int b  = pt / NPTS;
  int n  = pt - b * NPTS;
  const float* xb = xyz + (size_t)b * 3 * NPTS;
  float x = xb[n], y = xb[NPTS + n], z = xb[2 * NPTS + n];
  float p = sw1a[ch * 3 + 0] * x + sw1a[ch * 3 + 1] * y + sw1a[ch * 3 + 2] * z;
  float q = sw1d[ch * 3 + 0] * x + sw1d[ch * 3 + 1] * y + sw1d[ch * 3 + 2] * z + bias1p[ch];
  float* o = PQ + (size_t)pt * CH;
  o[ch] = p;
  o[HALF_CH + ch] = q;
}

// ---------------------------------------------------------------------------
// Kernel 4: feature GEMM, Cf[b][n][128] = Wc(128x128) @ F_b(128x2048), using
// V_WMMA_F32_16X16X4_F32 (exact fp32). One wave per 16x16 tile; 8 waves cover
// all 128 output channels of one 16-point tile. Operand layouts per ISA:
//   A (16x4): M = lane%16, K = vgpr + 2*(lane>=16)  -> contiguous float2 of Wc
//   B (4x16): N = lane%16, K = vgpr + 2*(lane>=16)  -> two strided row loads
//   D (16x16): N = lane%16, M = vgpr + 8*(lane>=16) -> 8 contiguous floats
// ---------------------------------------------------------------------------
__global__ void feat_gemm(const float* __restrict__ feat, const float* __restrict__ Wc,
                          float* __restrict__ Cf) {
  const int b    = blockIdx.y;
  const int n0   = blockIdx.x * 16;
  const int wave = threadIdx.x >> 5;
  const int lane = threadIdx.x & 31;
  const int l16  = lane & 15;
  const int hi   = lane >> 4;
  const int m0   = wave * 16;

  const float* Fb   = feat + (size_t)b * CH * NPTS;
  const float* arow = Wc + (size_t)(m0 + l16) * CH + 2 * hi;      // K offset 2*hi
  const float* bcol = Fb + (size_t)(2 * hi) * NPTS + n0 + l16;    // K offset 2*hi

  v8f acc = {0.f, 0.f, 0.f, 0.f, 0.f, 0.f, 0.f, 0.f};

  #pragma unroll 8
  for (int k = 0; k < CH; k += 4) {
    v2f a;
    a.x = arow[k];
    a.y = arow[k + 1];
    v2f bb;
    bb.x = bcol[(size_t)k * NPTS];
    bb.y = bcol[(size_t)(k + 1) * NPTS];
    acc = __builtin_amdgcn_wmma_f32_16x16x4_f32(false, a, false, bb,
                                                (short)0, acc, false, false);
  }

  float* out = Cf + ((size_t)b * NPTS + n0 + l16) * CH + m0 + 8 * hi;
  #pragma unroll
  for (int v = 0; v < 8; ++v) out[v] = acc[v];
}

// ---------------------------------------------------------------------------
// Kernel 5: fused gather + max + ReLU. Center terms are k-invariant, so max
// only the gathered neighbor terms; ReLU once at the end (monotone). Gather
// tables are 2 MB/batch -> L2 resident. 64 channel-threads read 256 B
// contiguous per neighbor.
// ---------------------------------------------------------------------------
__global__ void gather_max(const float* __restrict__ PQ, const float* __restrict__ Cf,
                           const int* __restrict__ idx, const float* __restrict__ bias2p,
                           float* __restrict__ out) {
  const int b  = blockIdx.y;
  const int ch = threadIdx.x & (HALF_CH - 1);
  const int n  = blockIdx.x * 4 + (threadIdx.x >> 6);

  const size_t pbase = (size_t)b * NPTS + n;
  const int* id = idx + pbase * KNN_K;

  const float qg = PQ[pbase * CH + HALF_CH + ch];
  const float vf = Cf[pbase * CH + HALF_CH + ch] + bias2p[ch];

  float mg = -FLT_MAX, mf = -FLT_MAX;
  #pragma unroll 4
  for (int k = 0; k < KNN_K; ++k) {
    int j = id[k];
    size_t jb = ((size_t)b * NPTS + j) * CH;
    mg = fmaxf(mg, PQ[jb + ch]);
    mf = fmaxf(mf, Cf[jb + ch]);
  }
  float geo = fmaxf(mg + qg, 0.0f);
  float fea = fmaxf(mf + vf, 0.0f);

  out[(size_t)b * CH * NPTS + (size_t)ch * NPTS + n]             = geo;
  out[(size_t)b * CH * NPTS + (size_t)(HALF_CH + ch) * NPTS + n] = fea;
}

// ---------------------------------------------------------------------------
extern "C" void kernel_launch(void* const* d_in, const int* in_sizes, int n_in,
                              void* d_out, int out_size, void* d_ws, size_t ws_size,
                              hipStream_t stream) {
  (void)in_sizes; (void)n_in; (void)out_size; (void)ws_size;

  const float* xyz  = (const float*)d_in[0];
  const float* feat = (const float*)d_in[1];
  const float* w1   = (const float*)d_in[2];
  const float* b1   = (const float*)d_in[3];
  const float* g1   = (const float*)d_in[4];
  const float* be1  = (const float*)d_in[5];
  const float* m1   = (const float*)d_in[6];
  const float* v1   = (const float*)d_in[7];
  const float* w2   = (const float*)d_in[8];
  const float* b2   = (const float*)d_in[9];
  const float* g2   = (const float*)d_in[10];
  const float* be2  = (const float*)d_in[11];
  const float* m2   = (const float*)d_in[12];
  const float* v2   = (const float*)d_in[13];

  // workspace layout (bytes): params (128 KB) | PQ (8 MB) | Cf (8 MB) | idx
  char* ws = (char*)d_ws;
  float* Wc     = (float*)ws;                 // 128*128 = 16384 floats
  float* sw1a   = Wc + 16384;                 // 192
  float* sw1d   = sw1a + 192;                 // 192
  float* bias1p = sw1d + 192;                 // 64
  float* bias2p = bias1p + 64;                // 64
  float* PQ     = (float*)(ws + (128 << 10));
  float* Cf     = PQ + (size_t)BATCH * NPTS * CH;
  int*   idxb   = (int*)(Cf + (size_t)BATCH * NPTS * CH);

  prep_params<<<1, 64, 0, stream>>>(w1, b1, g1, be1, m1, v1,
                                    w2, b2, g2, be2, m2, v2,
                                    Wc, sw1a, sw1d, bias1p, bias2p);

  knn_kernel<<<dim3(NPTS / 256, BATCH), 256, 0, stream>>>(xyz, idxb);

  geo_kernel<<<(BATCH * NPTS * HALF_CH) / 256, 256, 0, stream>>>(xyz, sw1a, sw1d,
                                                                 bias1p, PQ);

  feat_gemm<<<dim3(NPTS / 16, BATCH), 256, 0, stream>>>(feat, Wc, Cf);

  gather_max<<<dim3(NPTS / 4, BATCH), 256, 0, stream>>>(PQ, Cf, idxb, bias2p,
                                                        (float*)d_out);
}